// StackedMamba_54743653155145
// MI455X (gfx1250) — compile-verified
//
#include <hip/hip_runtime.h>
#include <math.h>

typedef __attribute__((ext_vector_type(2))) float v2f;
typedef __attribute__((ext_vector_type(8))) float v8f;

#define BATCH 2
#define SEQ   1024
#define DM    512          // d_model
#define DI    1024         // d_inner
#define DS    128          // d_state
#define DR    32           // dt_rank
#define NX    288          // DR + 2*DS
#define ROWS  (BATCH*SEQ)  // 2048

#define EPI_NONE     0
#define EPI_SOFTPLUS 1
#define EPI_RESID    2

// ---------------------------------------------------------------- RMSNorm
__global__ __launch_bounds__(128)
void rmsnorm_kernel(const float* __restrict__ x, const float* __restrict__ w,
                    float* __restrict__ out) {
  __shared__ float red[128];
  const int row = blockIdx.x;
  const float* xr = x + (size_t)row * DM;
  float s = 0.f;
  for (int i = threadIdx.x; i < DM; i += 128) { float v = xr[i]; s += v * v; }
  red[threadIdx.x] = s;
  __syncthreads();
  for (int o = 64; o > 0; o >>= 1) {
    if ((int)threadIdx.x < o) red[threadIdx.x] += red[threadIdx.x + o];
    __syncthreads();
  }
  const float rs = rsqrtf(red[0] * (1.f / DM) + 1e-5f);
  for (int i = threadIdx.x; i < DM; i += 128)
    out[(size_t)row * DM + i] = xr[i] * rs * w[i];
}

// ------------------------------------------------- generic fp32 WMMA GEMM
// C[row, col] = sum_k A[row, k] * Bw[col, k]   (+ epilogue)
// grid: (N/32, ROWS/64), block: 256 = 8 waves; each wave -> one 16x16 tile.
// Tiles are staged global->LDS with GLOBAL_LOAD_ASYNC_TO_LDS_B128 and
// double-buffered (software pipeline over the K chunks, ASYNCcnt-tracked).
__global__ __launch_bounds__(256)
void gemm_wmma_kernel(const float* __restrict__ A, int lda,
                      const float* __restrict__ Bw,
                      float* __restrict__ C, int N, int K,
                      int epi, const float* __restrict__ extra) {
  // stride 36 floats: rows stay 16B-aligned for B128, conflict-free mod 64 banks
  __shared__ __align__(16) float As[2][64][36];
  __shared__ __align__(16) float Bs[2][32][36];
  const int tid  = threadIdx.x;
  const int wave = tid >> 5;
  const int lane = tid & 31;
  const int m0 = blockIdx.y * 64;
  const int n0 = blockIdx.x * 32;
  const int wr = (wave >> 1) * 16;   // wave's row sub-tile: 0,16,32,48
  const int wc = (wave & 1) * 16;    // wave's col sub-tile: 0,16
  const int ml = lane & 15;          // M (for A) / N (for B) index
  const int kh = lane >> 4;          // K half select

  // staging assignment: each thread moves 3 x 16B per chunk
  const int rA = tid >> 3;           // 0..31
  const int cA = (tid & 7) * 4;      // 0,4,...,28

  auto issue = [&](int k0, int buf) {
    const float* gA0 = A + (size_t)(m0 + rA) * lda + (k0 + cA);
    const float* gA1 = A + (size_t)(m0 + rA + 32) * lda + (k0 + cA);
    const float* gB  = Bw + (size_t)(n0 + rA) * K + (k0 + cA);
    // low 32 bits of a generic shared pointer == LDS byte offset (aperture rule)
    unsigned lA0 = (unsigned)(size_t)&As[buf][rA][cA];
    unsigned lA1 = (unsigned)(size_t)&As[buf][rA + 32][cA];
    unsigned lB  = (unsigned)(size_t)&Bs[buf][rA][cA];
    asm volatile("global_load_async_to_lds_b128 %0, %1, off"
                 :: "v"(lA0), "v"(gA0) : "memory");
    asm volatile("global_load_async_to_lds_b128 %0, %1, off"
                 :: "v"(lA1), "v"(gA1) : "memory");
    asm volatile("global_load_async_to_lds_b128 %0, %1, off"
                 :: "v"(lB), "v"(gB) : "memory");
  };

  v8f acc = {};
  int buf = 0;
  issue(0, 0);
  for (int k0 = 0; k0 < K; k0 += 32) {
    if (k0 + 32 < K) {
      issue(k0 + 32, buf ^ 1);               // prefetch next chunk, other buffer
      asm volatile("s_wait_asynccnt 0x3" ::: "memory");  // current chunk landed
    } else {
      asm volatile("s_wait_asynccnt 0x0" ::: "memory");
    }
    __syncthreads();
#pragma unroll
    for (int kk = 0; kk < 32; kk += 4) {
      // ISA layout: lane = m + 16*(k>>1), vgpr = k&1  (A and B symmetric)
      v2f a, b;
      a.x = As[buf][wr + ml][kk + 2 * kh + 0];
      a.y = As[buf][wr + ml][kk + 2 * kh + 1];
      b.x = Bs[buf][wc + ml][kk + 2 * kh + 0];
      b.y = Bs[buf][wc + ml][kk + 2 * kh + 1];
      acc = __builtin_amdgcn_wmma_f32_16x16x4_f32(
          /*neg_a=*/false, a, /*neg_b=*/false, b,
          /*c_mod=*/(short)0, acc, /*reuse_a=*/false, /*reuse_b=*/false);
    }
    __syncthreads();                          // protect buf before next overwrite
    buf ^= 1;
  }
  // C/D layout: vgpr r -> M = r + 8*(lane>>4), N = lane&15
#pragma unroll
  for (int r = 0; r < 8; ++r) {
    const int row = m0 + wr + r + 8 * kh;
    const int col = n0 + wc + ml;
    float v = acc[r];
    if (epi == EPI_SOFTPLUS) {
      v += extra[col];
      v = (v > 20.f) ? v : log1pf(__expf(v));
    } else if (epi == EPI_RESID) {
      v += extra[(size_t)row * N + col];
    }
    C[(size_t)row * N + col] = v;
  }
}

// ----------------------------------------------- depthwise causal conv + SiLU
__global__ __launch_bounds__(256)
void conv_silu_kernel(const float* __restrict__ xz, const float* __restrict__ cw,
                      const float* __restrict__ cb, float* __restrict__ u) {
  const int idx = blockIdx.x * blockDim.x + threadIdx.x;   // ROWS*DI
  const int d   = idx & (DI - 1);
  const int row = idx >> 10;          // /DI
  const int b   = row >> 10;          // /SEQ
  const int t   = row & (SEQ - 1);
  float acc = cb[d];
#pragma unroll
  for (int j = 0; j < 4; ++j) {
    const int tt = t + j - 3;
    if (tt >= 0)
      acc += xz[(size_t)(b * SEQ + tt) * (2 * DI) + d] * cw[d * 4 + j];
  }
  u[(size_t)row * DI + d] = acc / (1.f + __expf(-acc));    // silu
}

// ----------------------------------------------------------- selective scan
// one wave32 per (b, d) channel; lane holds 4 of the 128 states.
__global__ __launch_bounds__(256)
void scan_kernel(const float* __restrict__ dt, const float* __restrict__ u,
                 const float* __restrict__ dbl, const float* __restrict__ xz,
                 const float* __restrict__ A_log, const float* __restrict__ Dp,
                 float* __restrict__ y) {
  const int gwave = (blockIdx.x * blockDim.x + threadIdx.x) >> 5;
  const int lane  = threadIdx.x & 31;
  const int b = gwave >> 10;          // /DI
  const int d = gwave & (DI - 1);
  const size_t abase = (size_t)d * DS + 4 * lane;
  const float A0 = -__expf(A_log[abase + 0]);
  const float A1 = -__expf(A_log[abase + 1]);
  const float A2 = -__expf(A_log[abase + 2]);
  const float A3 = -__expf(A_log[abase + 3]);
  const float Dd = Dp[d];
  float h0 = 0.f, h1 = 0.f, h2 = 0.f, h3 = 0.f;
  for (int t = 0; t < SEQ; ++t) {
    const size_t row = (size_t)b * SEQ + t;
    if (t + 1 < SEQ) {   // pull next timestep's B/C row while we compute
      __builtin_prefetch(dbl + (row + 1) * NX + DR + 4 * lane, 0, 0);
      __builtin_prefetch(dbl + (row + 1) * NX + DR + DS + 4 * lane, 0, 0);
    }
    const float dtv = dt[row * DI + d];
    const float uv  = u[row * DI + d];
    const float4 Bv = *(reinterpret_cast<const float4*>(dbl + row * NX + DR) + lane);
    const float4 Cv = *(reinterpret_cast<const float4*>(dbl + row * NX + DR + DS) + lane);
    const float du = dtv * uv;
    h0 = __expf(dtv * A0) * h0 + du * Bv.x;
    h1 = __expf(dtv * A1) * h1 + du * Bv.y;
    h2 = __expf(dtv * A2) * h2 + du * Bv.z;
    h3 = __expf(dtv * A3) * h3 + du * Bv.w;
    float acc = h0 * Cv.x + h1 * Cv.y + h2 * Cv.z + h3 * Cv.w;
#pragma unroll
    for (int off = 16; off > 0; off >>= 1)
      acc += __shfl_xor(acc, off, 32);
    if (lane == 0) {
      const float z = xz[row * (2 * DI) + DI + d];
      y[row * DI + d] = (acc + uv * Dd) * (z / (1.f + __expf(-z)));
    }
  }
}

// ------------------------------------------------------------------- host
extern "C" void kernel_launch(void* const* d_in, const int* in_sizes, int n_in,
                              void* d_out, int out_size, void* d_ws, size_t ws_size,
                              hipStream_t stream) {
  const float* x      = (const float*)d_in[0];
  const float* norm_w = (const float*)d_in[1];
  const float* in_w   = (const float*)d_in[2];
  const float* conv_w = (const float*)d_in[3];
  const float* conv_b = (const float*)d_in[4];
  const float* xp_w   = (const float*)d_in[5];
  const float* dt_w   = (const float*)d_in[6];
  const float* dt_b   = (const float*)d_in[7];
  const float* A_log  = (const float*)d_in[8];
  const float* Dp     = (const float*)d_in[9];
  const float* out_w  = (const float*)d_in[10];
  float* out = (float*)d_out;

  float* base = (float*)d_ws;
  float* xcur = base;                               // ROWS*DM
  float* xn   = xcur + (size_t)ROWS * DM;           // ROWS*DM
  float* xz   = xn   + (size_t)ROWS * DM;           // ROWS*2*DI
  float* u    = xz   + (size_t)ROWS * 2 * DI;       // ROWS*DI
  float* dbl  = u    + (size_t)ROWS * DI;           // ROWS*NX
  float* dtb  = dbl  + (size_t)ROWS * NX;           // ROWS*DI
  float* y    = dtb  + (size_t)ROWS * DI;           // ROWS*DI

  for (int l = 0; l < 2; ++l) {
    const float* xin  = (l == 0) ? x : xcur;
    float*       xout = (l == 0) ? xcur : out;
    const float* nw  = norm_w + (size_t)l * DM;
    const float* iw  = in_w   + (size_t)l * 2 * DI * DM;
    const float* cw  = conv_w + (size_t)l * DI * 4;
    const float* cb  = conv_b + (size_t)l * DI;
    const float* xpw = xp_w   + (size_t)l * NX * DI;
    const float* dtw = dt_w   + (size_t)l * DI * DR;
    const float* dtbias = dt_b + (size_t)l * DI;
    const float* Al  = A_log  + (size_t)l * DI * DS;
    const float* Dl  = Dp     + (size_t)l * DI;
    const float* ow  = out_w  + (size_t)l * DM * DI;

    rmsnorm_kernel<<<ROWS, 128, 0, stream>>>(xin, nw, xn);

    gemm_wmma_kernel<<<dim3((2 * DI) / 32, ROWS / 64), 256, 0, stream>>>(
        xn, DM, iw, xz, 2 * DI, DM, EPI_NONE, nullptr);

    conv_silu_kernel<<<(ROWS * DI) / 256, 256, 0, stream>>>(xz, cw, cb, u);

    gemm_wmma_kernel<<<dim3(NX / 32, ROWS / 64), 256, 0, stream>>>(
        u, DI, xpw, dbl, NX, DI, EPI_NONE, nullptr);

    gemm_wmma_kernel<<<dim3(DI / 32, ROWS / 64), 256, 0, stream>>>(
        dbl, NX, dtw, dtb, DI, DR, EPI_SOFTPLUS, dtbias);

    scan_kernel<<<(BATCH * DI) / 8, 256, 0, stream>>>(dtb, u, dbl, xz, Al, Dl, y);

    gemm_wmma_kernel<<<dim3(DM / 32, ROWS / 64), 256, 0, stream>>>(
        y, DI, ow, xout, DM, DI, EPI_RESID, xin);
  }
}